// SimpleNEAgent_76338748719219
// MI455X (gfx1250) — compile-verified
//
#include <hip/hip_runtime.h>

#define IN_SIZE  256
#define N_NODES  1024
#define DEG      32
#define BATCH    16384
#define OUT_SIZE 16
#define NPOS     (IN_SIZE + N_NODES)      // 1280 activation slots actually used
#define TILE     32                        // batch columns resident per workgroup
#define BLOCK    128                       // 4 waves: 8 cols/wave x 4 k-groups
#define LDS_BYTES (NPOS * TILE * 4)        // 163840 B -> 2 workgroups per WGP (320KB)

#if defined(__has_builtin)
#if __has_builtin(__builtin_amdgcn_global_load_async_to_lds_b32) && \
    __has_builtin(__builtin_amdgcn_s_wait_asynccnt)
#define USE_ASYNC_LDS 1
#endif
#endif
#ifndef USE_ASYNC_LDS
#define USE_ASYNC_LDS 0
#endif

__global__ __launch_bounds__(BLOCK)
void neat_forward(const float* __restrict__ x,
                  const float* __restrict__ wts,
                  const int*   __restrict__ idxs,
                  float*       __restrict__ out)
{
    extern __shared__ float act[];         // [NPOS][TILE], position-major

    const int tid  = threadIdx.x;
    const int lane = tid & 31;
    const int wv   = tid >> 5;             // wave id in block: 0..3
    const int kg   = lane >> 3;            // k-group 0..3 (8 MACs each)
    const int col  = wv * 8 + (lane & 7);  // owned batch column 0..31
    const int b0   = blockIdx.x * TILE;    // global batch base

    // ---- Stage x[b0..b0+TILE)[0..255] transposed into LDS ----
    // flat f: p = f&255 (fast global dim, coalesced), b = f>>8
    {
        const float* xg = x + (size_t)b0 * IN_SIZE;
#if USE_ASYNC_LDS
        typedef __attribute__((address_space(1))) int gi_t;   // global (AS1)
        typedef __attribute__((address_space(3))) int li_t;   // LDS    (AS3)
        for (int f = tid; f < IN_SIZE * TILE; f += BLOCK) {
            const int p = f & (IN_SIZE - 1);
            const int b = f >> 8;
            __builtin_amdgcn_global_load_async_to_lds_b32(
                (gi_t*)(xg + (size_t)b * IN_SIZE + p),
                (li_t*)&act[p * TILE + b],
                0, 0);
        }
        __builtin_amdgcn_s_wait_asynccnt(0);
#else
        for (int f = tid; f < IN_SIZE * TILE; f += BLOCK) {
            const int p = f & (IN_SIZE - 1);
            const int b = f >> 8;
            act[p * TILE + b] = xg[(size_t)b * IN_SIZE + p];
        }
#endif
        __syncthreads();   // one-time: staging is cooperative across waves
    }

    // ---- Sequential node loop: each wave is fully independent (no barriers) ----
    const float* wlane = wts  + kg * 8;    // this lane's 8-wide k slice
    const int*   ilane = idxs + kg * 8;

    float4 w0 = *(const float4*)(wlane);
    float4 w1 = *(const float4*)(wlane + 4);
    int4   i0 = *(const int4*)(ilane);
    int4   i1 = *(const int4*)(ilane + 4);

    for (int node = 0; node < N_NODES; ++node) {
        // software pipeline: fetch node+1 operands while gathering node's inputs
        const int nn = (node + 1 < N_NODES) ? node + 1 : node;
        const float4 nw0 = *(const float4*)(wlane + (size_t)nn * DEG);
        const float4 nw1 = *(const float4*)(wlane + (size_t)nn * DEG + 4);
        const int4   ni0 = *(const int4*)(ilane + (size_t)nn * DEG);
        const int4   ni1 = *(const int4*)(ilane + (size_t)nn * DEG + 4);

        // prefetch a few nodes ahead (L2-resident after first touch)
        const int pf = (node + 8 < N_NODES) ? node + 8 : node;
        __builtin_prefetch(wts  + (size_t)pf * DEG, 0, 3);
        __builtin_prefetch(idxs + (size_t)pf * DEG, 0, 3);

        // 8 gathered MACs for this (column, k-group)
        float acc;
        acc = act[i0.x * TILE + col] * w0.x;
        acc = fmaf(act[i0.y * TILE + col], w0.y, acc);
        acc = fmaf(act[i0.z * TILE + col], w0.z, acc);
        acc = fmaf(act[i0.w * TILE + col], w0.w, acc);
        acc = fmaf(act[i1.x * TILE + col], w1.x, acc);
        acc = fmaf(act[i1.y * TILE + col], w1.y, acc);
        acc = fmaf(act[i1.z * TILE + col], w1.z, acc);
        acc = fmaf(act[i1.w * TILE + col], w1.w, acc);

        // reduce the 4 k-groups (lanes xor 8, xor 16) — wave-internal, no barrier
        acc += __shfl_xor(acc, 8, 32);
        acc += __shfl_xor(acc, 16, 32);

        if (kg == 0) {
            const float t = tanhf(acc);
            act[(IN_SIZE + node) * TILE + col] = t;   // visible to same wave, in-order DS
            if (node >= N_NODES - OUT_SIZE)
                out[(size_t)(node - (N_NODES - OUT_SIZE)) * BATCH + b0 + col] = t;
        }
        // keep the activation store ordered before next node's LDS gathers
        __asm__ volatile("" ::: "memory");

        w0 = nw0; w1 = nw1; i0 = ni0; i1 = ni1;
    }
}

extern "C" void kernel_launch(void* const* d_in, const int* in_sizes, int n_in,
                              void* d_out, int out_size, void* d_ws, size_t ws_size,
                              hipStream_t stream) {
    (void)in_sizes; (void)n_in; (void)d_ws; (void)ws_size; (void)out_size;
    const float* x   = (const float*)d_in[0];
    const float* w   = (const float*)d_in[1];
    const int*   idx = (const int*)d_in[2];
    float*       out = (float*)d_out;

    dim3 grid(BATCH / TILE);   // 512 workgroups
    dim3 block(BLOCK);
    hipLaunchKernelGGL(neat_forward, grid, block, LDS_BYTES, stream, x, w, idx, out);
}